// SmoothnessConstraint_36515811950785
// MI455X (gfx1250) — compile-verified
//
#include <hip/hip_runtime.h>

// actions: [B, T, A] fp32, T=256, A=64.
// out[b,0,a] = in[b,0,a]
// out[b,t,a] = out[b,t-1,a] + clamp(in[b,t,a]-in[b,t-1,a], -0.5, 0.5)
//
// Memory-bound streaming scan (536 MB -> ~23 us @ 23.3 TB/s). Input streams
// through LDS via gfx1250 GLOBAL_LOAD_ASYNC_TO_LDS_B128 (ASYNCcnt-tracked,
// quad-buffered, NT temporal hint); output leaves with NT b128 stores.
// One thread owns one (b, float4-slice-of-A) scan chain.

typedef float fvec4 __attribute__((ext_vector_type(4)));

#define T_DIM  256
#define A_DIM  64
#define AV     16                  // float4 slices per row
#define CHUNK  8                   // timesteps per pipeline stage
#define ROWS   8                   // b-rows per block
#define BLOCK  (ROWS * AV)         // 128 threads
#define NCHUNK (T_DIM / CHUNK)     // 32 stages
#define NBUF   4                   // pipeline depth (quad buffer)
#define MAXD   0.5f

static __device__ __forceinline__ fvec4 clamp4(fvec4 d) {
    // fmin/fmax pair -> v_med3_f32 per component
    fvec4 r;
    r.x = fminf(MAXD, fmaxf(-MAXD, d.x));
    r.y = fminf(MAXD, fmaxf(-MAXD, d.y));
    r.z = fminf(MAXD, fmaxf(-MAXD, d.z));
    r.w = fminf(MAXD, fmaxf(-MAXD, d.w));
    return r;
}

// s_wait_asynccnt with a literal immediate
#define WAIT_ASYNC(N) asm volatile("s_wait_asynccnt " #N ::: "memory")
#define WAIT_DS0()    asm volatile("s_wait_dscnt 0x0" ::: "memory")

// One async lane-transfer: IOFFSET is applied by HW to BOTH the LDS and the
// global address; our LDS timestep stride (256 B) matches the global one, so
// a single base pair serves the whole chunk.
#define ASYNC_LD(OFS)                                                        \
    asm volatile("global_load_async_to_lds_b128 %0, %1, off offset:" #OFS    \
                 " th:TH_LOAD_NT" :: "v"(la), "v"(ga) : "memory")

__global__ __launch_bounds__(BLOCK) void smooth_scan_kernel(
    const float* __restrict__ in, float* __restrict__ out) {
#if defined(__gfx1250__)
    // Staging: [buf][row][step][v], 16 B elements -> 4 x 16 KB = 64 KB.
    __shared__ fvec4 sbuf[NBUF][ROWS][CHUNK][AV];

    const int tid = threadIdx.x;
    const int v   = tid & (AV - 1);
    const int row = tid >> 4;
    const size_t b = (size_t)blockIdx.x * ROWS + row;

    const float* __restrict__ gbase = in  + b * (T_DIM * A_DIM) + (size_t)v * 4;
    float*       __restrict__ obase = out + b * (T_DIM * A_DIM) + (size_t)v * 4;

    // Per-thread LDS base byte address of buffer 0 (low 32 bits of generic
    // pointer == LDS offset); buffers are 16 KB apart.
    const unsigned la0 = (unsigned)(size_t)&sbuf[0][row][0][v];

    // Issue one chunk: CHUNK async b128 lane-transfers into sbuf[c % NBUF].
    auto issue = [&](int c) {
        unsigned long long ga =
            (unsigned long long)(gbase + (size_t)c * (CHUNK * A_DIM));
        unsigned la = la0 + ((unsigned)(c & (NBUF - 1)) << 14);
        ASYNC_LD(0);    ASYNC_LD(256);  ASYNC_LD(512);  ASYNC_LD(768);
        ASYNC_LD(1024); ASYNC_LD(1280); ASYNC_LD(1536); ASYNC_LD(1792);
    };

    fvec4 prev, acc;

    // Read chunk c back from LDS, advance the scan, NT-store the outputs.
    auto process = [&](int c, bool first) {
        fvec4 cur[CHUNK];
        #pragma unroll
        for (int k = 0; k < CHUNK; ++k)
            cur[k] = sbuf[c & (NBUF - 1)][row][k][v];
        float* op = obase + (size_t)c * (CHUNK * A_DIM);
        #pragma unroll
        for (int k = 0; k < CHUNK; ++k) {
            if (first && k == 0) {
                acc  = cur[0];
                prev = cur[0];
                __builtin_nontemporal_store(acc, (fvec4*)op);
            } else {
                fvec4 d = clamp4(cur[k] - prev);
                acc  = acc + d;
                prev = cur[k];
                __builtin_nontemporal_store(acc, (fvec4*)(op + (size_t)k * A_DIM));
            }
        }
    };

    // ---- software pipeline: depth-4 async buffering ----
    issue(0); issue(1); issue(2); issue(3);        // 32 transfers in flight

    WAIT_ASYNC(24);                // chunk 0 landed (<= 3 chunks outstanding)
    process(0, true);
    WAIT_DS0();                    // readback drained before refilling buf 0
    issue(4);

    for (int c = 1; c <= NCHUNK - 5; ++c) {        // c = 1..27
        // warm L2 six chunks ahead (clamped; speculative prefetch is safe)
        int cp = c + 6;
        cp = (cp < NCHUNK) ? cp : (NCHUNK - 1);
        __builtin_prefetch((const void*)(gbase + (size_t)cp * (CHUNK * A_DIM)), 0, 0);

        WAIT_ASYNC(24);            // chunk c landed
        process(c, false);
        WAIT_DS0();                // buffer reads done
        issue(c + 4);              // refill buf[c & 3]
    }

    // ---- drain tail: chunks 28..31 ----
    WAIT_ASYNC(24); process(NCHUNK - 4, false);
    WAIT_ASYNC(16); process(NCHUNK - 3, false);
    WAIT_ASYNC(8);  process(NCHUNK - 2, false);
    WAIT_ASYNC(0);  process(NCHUNK - 1, false);
#else
    // Non-gfx1250 fallback: direct-load sequential scan (same mapping).
    const int tid = blockIdx.x * BLOCK + threadIdx.x;
    const int v   = tid & (AV - 1);
    const size_t b = (size_t)(tid >> 4);
    const fvec4* pin  = (const fvec4*)(in  + b * (T_DIM * A_DIM) + (size_t)v * 4);
    fvec4*       pout = (fvec4*)      (out + b * (T_DIM * A_DIM) + (size_t)v * 4);
    fvec4 prev = pin[0];
    fvec4 acc  = prev;
    pout[0] = acc;
    for (int t = 1; t < T_DIM; ++t) {
        fvec4 cur = pin[(size_t)t * AV];
        fvec4 d   = clamp4(cur - prev);
        acc  = acc + d;
        prev = cur;
        pout[(size_t)t * AV] = acc;
    }
#endif
}

extern "C" void kernel_launch(void* const* d_in, const int* in_sizes, int n_in,
                              void* d_out, int out_size, void* d_ws, size_t ws_size,
                              hipStream_t stream) {
    (void)n_in; (void)d_ws; (void)ws_size; (void)out_size;
    const float* in  = (const float*)d_in[0];
    float*       out = (float*)d_out;

    const int B = in_sizes[0] / (T_DIM * A_DIM);   // 4096 for the reference
    const int grid = B / ROWS;                     // 512 blocks of 128 threads
    smooth_scan_kernel<<<grid, BLOCK, 0, stream>>>(in, out);
}